// SequenceFoam_83794811945124
// MI455X (gfx1250) — compile-verified
//
#include <hip/hip_runtime.h>
#include <hip/hip_bf16.h>
#include <math.h>

#define D        256
#define NB       256
#define SEQ_LEN  512
#define N_EQ     5
#define ETA_C    0.1f
#define KAPPA_C  0.05f
#define NSWEEPS  6
#define NTHREADS 1024
#define ASTRIDE  257          // odd row stride -> conflict-free LDS column access
#define USE_TDM  1

typedef __attribute__((ext_vector_type(2))) float v2f;
typedef __attribute__((ext_vector_type(8))) float v8f;
typedef unsigned int u32x4 __attribute__((ext_vector_type(4)));
typedef int i32x4 __attribute__((ext_vector_type(4)));
typedef int i32x8 __attribute__((ext_vector_type(8)));

// Single persistent workgroup (32 wave32 waves) walks the 512-token scan.
// S and MEM live in d_ws (L2-resident, 512 KB); rho lives in LDS (263 KB,
// padded stride) so the Jacobi eigensolver -- the dominant cost -- runs on
// LDS bandwidth instead of VMEM.
__global__ __launch_bounds__(NTHREADS, 1)
void foam_seq_kernel(const float* __restrict__ embed,
                     const float* __restrict__ bubbles,
                     const float* __restrict__ base_p,
                     const float* __restrict__ sens_p,
                     const int*   __restrict__ tokens,
                     float* __restrict__ out,
                     float* __restrict__ ws)
{
    __shared__ float Alds[D * ASTRIDE];       // rho, Jacobi in place (263 KB)
    __shared__ float red[NTHREADS];
    __shared__ float xbuf[D], mmean[D], cmean[D], xs[D];
    __shared__ float invn[NB];
    __shared__ float rc[128], rs[128];
    __shared__ int   rp[128], rq[128];
    __shared__ float wbuf[D];
    __shared__ float scal[2];
    __shared__ int   toksh;

    float* __restrict__ S   = ws;               // [NB][D] bubble states
    float* __restrict__ MEM = ws + NB * D;      // [NB][D] memory state

    const int tid  = threadIdx.x;
    const int lane = tid & 31;   // wave32
    const int wave = tid >> 5;   // 32 waves

    for (int i = tid; i < NB * D; i += NTHREADS) MEM[i] = 0.0f;
    __syncthreads();

    const float base = base_p[0];
    const float sens = fabsf(sens_p[0]);

    for (int step = 0; step < SEQ_LEN; ++step) {
        if (tid == 0) toksh = tokens[step];
        __syncthreads();
        const int tok = toksh;

#if USE_TDM
        // ---- async TDM load of embed row -> xbuf (overlaps mem-mean) ----
        if (tid == 0) {
            const unsigned long long ga =
                (unsigned long long)(const void*)(embed + (size_t)tok * D);
            const unsigned lds_off = (unsigned)(unsigned long long)(void*)&xbuf[0];
            u32x4 g0;
            g0[0] = 1u;                                   // count=1, user desc
            g0[1] = lds_off;                              // lds_addr
            g0[2] = (unsigned)(ga & 0xFFFFFFFFu);         // global_addr[31:0]
            g0[3] = (unsigned)((ga >> 32) & 0x1FFFFFFu)   // global_addr[56:32]
                    | 0x80000000u;                        // type=2 ("image")
            i32x8 g1;
            g1[0] = 0x20000;          // data_size=2 (4B), workgroup_mask=0
            g1[1] = (int)(256u << 16);// tensor_dim0[15:0] = 256
            g1[2] = (int)(1u << 16);  // tensor_dim0 hi=0, tensor_dim1 lo=1
            g1[3] = (int)(256u << 16);// tensor_dim1 hi=0, tile_dim0=256
            g1[4] = 1;                // tile_dim1=1, tile_dim2=0
            g1[5] = 256;              // tensor_dim0_stride = 256
            g1[6] = (int)(256u << 16);// tensor_dim1_stride lo16 (unused row)
            g1[7] = 0;
            i32x4 g2 = {0, 0, 0, 0};
            i32x4 g3 = {0, 0, 0, 0};
#if defined(__clang_major__) && (__clang_major__ >= 23)
            i32x8 g4 = {0, 0, 0, 0, 0, 0, 0, 0};
            __builtin_amdgcn_tensor_load_to_lds(g0, g1, g2, g3, g4, 0);
#else
            __builtin_amdgcn_tensor_load_to_lds(g0, g1, g2, g3, 0);
#endif
        }
#endif
        // prefetch next token's embedding row (global_prefetch_b8)
        if (step + 1 < SEQ_LEN && tid < 8) {
            const int nt = tokens[step + 1];
            __builtin_prefetch(embed + (size_t)nt * D + tid * 32, 0, 0);
        }

        // ---- mem_mean (column mean of MEM) ----
        {
            const int d  = tid & (D - 1);
            const int rb = tid >> 8;
            const float* mp = MEM + rb * 64 * D + d;
            float acc = 0.0f;
            for (int r = 0; r < 64; ++r) acc += mp[r * D];
            red[tid] = acc;
        }
#if USE_TDM
        if (tid == 0) __builtin_amdgcn_s_wait_tensorcnt(0);
#else
        if (tid < D) xbuf[tid] = embed[(size_t)tok * D + tid];
#endif
        __syncthreads();
        if (tid < D) {
            const float m = (red[tid] + red[tid + 256] + red[tid + 512] + red[tid + 768]) * (1.0f / NB);
            mmean[tid] = m;
            const float xv = xbuf[tid];
            red[tid]       = xv * m;
            red[tid + 256] = xv * xv;
            red[tid + 512] = m * m;
        }
        __syncthreads();
        if (tid == 0) {
            float dxm = 0.0f, x2 = 0.0f, m2 = 0.0f;
            for (int i = 0; i < D; ++i) { dxm += red[i]; x2 += red[i + 256]; m2 += red[i + 512]; }
            const float mem_norm = sqrtf(m2) + 1e-10f;
            const float x_norm   = sqrtf(x2) + 1e-10f;
            const float novelty  = (mem_norm > 1e-8f) ? (1.0f - dxm / (x_norm * mem_norm)) : 1.0f;
            const float z = base - sens * novelty;
            scal[0] = 1.0f / (1.0f + __expf(-z));
        }
        __syncthreads();
        const float decay = scal[0];
        if (tid < D) xs[tid] = xbuf[tid] + decay * mmean[tid];

        // ---- s = bubbles ----
        for (int i = tid; i < NB * D; i += NTHREADS) S[i] = bubbles[i];
        __syncthreads();

        // ---- 5 equilibration iterations ----
        for (int it = 0; it < N_EQ; ++it) {
            {   // column mean of S
                const int d  = tid & (D - 1);
                const int rb = tid >> 8;
                const float* sp = S + rb * 64 * D + d;
                float acc = 0.0f;
                for (int r = 0; r < 64; ++r) acc += sp[r * D];
                red[tid] = acc;
            }
            __syncthreads();
            if (tid < D)
                cmean[tid] = (red[tid] + red[tid + 256] + red[tid + 512] + red[tid + 768]) * (1.0f / NB);
            __syncthreads();
            {   // relax + row sumsq (4 threads per row)
                const int r  = tid >> 2;
                const int l4 = tid & 3;
                float* sr = S + r * D;
                float sq = 0.0f;
                for (int jj = 0; jj < 64; ++jj) {
                    const int j = l4 + 4 * jj;
                    const float sv = sr[j];
                    const float nv = sv + ETA_C * (xs[j] - sv) + KAPPA_C * (cmean[j] - sv);
                    sr[j] = nv;
                    sq += nv * nv;
                }
                red[tid] = sq;
            }
            __syncthreads();
            if (tid < NB) {
                const float t4 = red[4 * tid] + red[4 * tid + 1] + red[4 * tid + 2] + red[4 * tid + 3];
                invn[tid] = 1.0f / (sqrtf(t4) + 1e-10f);
            }
            __syncthreads();
            {
                const int r  = tid >> 2;
                const int l4 = tid & 3;
                float* sr = S + r * D;
                const float inv = invn[r];
                for (int jj = 0; jj < 64; ++jj) sr[l4 + 4 * jj] *= inv;
            }
            __syncthreads();
        }

        // ---- rho = S^T S / NB : WMMA fp32, 2x4 tile block per wave ----
        // 8 independent accumulator chains per wave; 12 loads feed 8 WMMAs.
        {
            const int m   = lane & 15;
            const int kh  = (lane >> 4) << 1;     // 0 or 2
            const int ti0 = (wave >> 2) << 1;     // 0..14 step 2
            const int tj0 = (wave & 3) << 2;      // 0,4,8,12
            const float* cA0 = S + 16 * ti0 + m;
            const float* cA1 = cA0 + 16;
            const float* cB  = S + 16 * tj0 + m;
            const v8f vz = {0.0f, 0.0f, 0.0f, 0.0f, 0.0f, 0.0f, 0.0f, 0.0f};
            v8f acc[2][4];
            #pragma unroll
            for (int i = 0; i < 2; ++i)
                #pragma unroll
                for (int j = 0; j < 4; ++j) acc[i][j] = vz;

            for (int k0 = 0; k0 < NB; k0 += 4) {
                const int off0 = (k0 + kh) * D;
                const int off1 = off0 + D;
                v2f a0, a1, b[4];
                a0.x = cA0[off0]; a0.y = cA0[off1];
                a1.x = cA1[off0]; a1.y = cA1[off1];
                #pragma unroll
                for (int j = 0; j < 4; ++j) {
                    b[j].x = cB[off0 + 16 * j];
                    b[j].y = cB[off1 + 16 * j];
                }
                #pragma unroll
                for (int j = 0; j < 4; ++j) {
                    acc[0][j] = __builtin_amdgcn_wmma_f32_16x16x4_f32(
                        false, a0, false, b[j], (short)0, acc[0][j], false, false);
                    acc[1][j] = __builtin_amdgcn_wmma_f32_16x16x4_f32(
                        false, a1, false, b[j], (short)0, acc[1][j], false, false);
                }
            }
            const int li = lane & 15;
            const int hi = (lane >> 4) << 3;
            #pragma unroll
            for (int i = 0; i < 2; ++i)
                #pragma unroll
                for (int j = 0; j < 4; ++j) {
                    float* orow = Alds + (16 * (ti0 + i) + hi) * ASTRIDE + 16 * (tj0 + j) + li;
                    #pragma unroll
                    for (int vv = 0; vv < 8; ++vv)
                        orow[vv * ASTRIDE] = acc[i][j][vv] * (1.0f / NB);
                }
        }
        __syncthreads();

        // ---- eigenvalues: parallel cyclic Jacobi in LDS ----
        for (int sweep = 0; sweep < NSWEEPS; ++sweep) {
            for (int rr = 0; rr < NB - 1; ++rr) {
                if (tid < 128) {   // 128 disjoint rotations (tournament)
                    const int k = tid;
                    int p = (k == 0) ? 0 : ((k - 1 + rr) % (NB - 1)) + 1;
                    int q = ((NB - 2 - k + rr) % (NB - 1)) + 1;
                    if (p > q) { const int t = p; p = q; q = t; }
                    const float app = Alds[p * ASTRIDE + p];
                    const float aqq = Alds[q * ASTRIDE + q];
                    const float apq = Alds[p * ASTRIDE + q];
                    float c = 1.0f, s = 0.0f;
                    if (fabsf(apq) > 1e-20f) {
                        const float tau = (aqq - app) / (2.0f * apq);
                        const float t = copysignf(1.0f, tau) / (fabsf(tau) + sqrtf(1.0f + tau * tau));
                        c = rsqrtf(1.0f + t * t);
                        s = t * c;
                    }
                    rp[tid] = p; rq[tid] = q; rc[tid] = c; rs[tid] = s;
                }
                __syncthreads();
                {   // row phase: 8 threads per pair
                    const int mi = tid >> 3, l8 = tid & 7;
                    const int p = rp[mi], q = rq[mi];
                    const float c = rc[mi], s = rs[mi];
                    float* Ap = Alds + p * ASTRIDE;
                    float* Aq = Alds + q * ASTRIDE;
                    for (int jj = 0; jj < 32; ++jj) {
                        const int j = l8 + 8 * jj;
                        const float ap = Ap[j], aq = Aq[j];
                        Ap[j] = c * ap - s * aq;
                        Aq[j] = s * ap + c * aq;
                    }
                }
                __syncthreads();
                {   // column phase (odd ASTRIDE -> no bank conflicts)
                    const int mi = tid >> 3, l8 = tid & 7;
                    const int p = rp[mi], q = rq[mi];
                    const float c = rc[mi], s = rs[mi];
                    for (int ii = 0; ii < 32; ++ii) {
                        const int i = l8 + 8 * ii;
                        const float ap = Alds[i * ASTRIDE + p], aq = Alds[i * ASTRIDE + q];
                        Alds[i * ASTRIDE + p] = c * ap - s * aq;
                        Alds[i * ASTRIDE + q] = s * ap + c * aq;
                    }
                }
                __syncthreads();
            }
        }

        // ---- spectrum: clip, ascending rank-sort, normalize ----
        if (tid < D) wbuf[tid] = fmaxf(Alds[tid * ASTRIDE + tid], 0.0f);
        __syncthreads();
        if (tid == 0) {
            float sum = 0.0f;
            for (int i = 0; i < D; ++i) sum += wbuf[i];
            scal[1] = 1.0f / (sum + 1e-10f);
        }
        __syncthreads();
        if (tid < D) {
            const float wv = wbuf[tid];
            int rank = 0;
            for (int j = 0; j < D; ++j) {
                const float wj = wbuf[j];
                rank += (wj < wv || (wj == wv && j < tid)) ? 1 : 0;
            }
            out[(size_t)step * D + rank] = wv * scal[1];
        }

        // ---- memory = decay*memory + (1-decay)*S ----
        for (int i = tid; i < NB * D; i += NTHREADS)
            MEM[i] = decay * MEM[i] + (1.0f - decay) * S[i];
        __syncthreads();
    }
}

extern "C" void kernel_launch(void* const* d_in, const int* in_sizes, int n_in,
                              void* d_out, int out_size, void* d_ws, size_t ws_size,
                              hipStream_t stream) {
    (void)in_sizes; (void)n_in; (void)out_size; (void)ws_size;
    const float* embed   = (const float*)d_in[0];
    const float* bubbles = (const float*)d_in[1];
    const float* base    = (const float*)d_in[2];
    const float* sens    = (const float*)d_in[3];
    const int*   tokens  = (const int*)d_in[4];
    float* out = (float*)d_out;
    float* ws  = (float*)d_ws;   // needs 2*256*256*4 = 512 KB

    foam_seq_kernel<<<dim3(1), dim3(NTHREADS), 0, stream>>>(
        embed, bubbles, base, sens, tokens, out, ws);
}